// LA_283467842715
// MI455X (gfx1250) — compile-verified
//
#include <hip/hip_runtime.h>
#include <hip/hip_bf16.h>

// Problem constants (fixed by the reference):
//   x: [B=256, C=2048, H=8, W=8] f32, HEAD=16, c16=128.
// Bandwidth-bound: >= 268 MB traffic -> ~11.5us at 23.3 TB/s. Strategy:
//   K1: stream-read x once via TDM (tensor_load_to_lds) into LDS, reduce to
//       means/maxs [B,16,4,4] in d_ws (deterministic fixed-order reduction).
//   K2: per-(b,head) WG: cross-head MLP via V_WMMA_F32_16X16X4_F32, bilinear
//       4x4->8x8 gate, out = x*(1+sigmoid(gate)) with float4 streaming
//       (x re-read hits the 192MB L2 since x is only 128MB).

typedef float v2f __attribute__((ext_vector_type(2)));
typedef float v8f __attribute__((ext_vector_type(8)));
typedef unsigned v4u __attribute__((ext_vector_type(4)));
typedef int v4i __attribute__((ext_vector_type(4)));
typedef int v8i __attribute__((ext_vector_type(8)));

#define CHUNK_FLOATS 8192   // 128 channels * 64 pixels = 32 KiB

#if __has_builtin(__builtin_amdgcn_tensor_load_to_lds)
#define STAGE_MODE 2
#elif __has_builtin(__builtin_amdgcn_global_load_async_to_lds_b128)
#define STAGE_MODE 1
#else
#define STAGE_MODE 0
#endif

#if STAGE_MODE == 2
// ---- TDM path: one tensor_load_to_lds per 32 KiB chunk (issued by wave 0) ----
__device__ __forceinline__ void stage_issue(const float* gsrc, float* lds, int t) {
  if (t < 32) {
    const unsigned long long ga = (unsigned long long)gsrc;
    const unsigned la = (unsigned)(unsigned long long)lds;
    v4u g0;
    g0.x = 1u;                                         // count=1, user desc, no gather
    g0.y = la;                                         // lds_addr (bytes)
    g0.z = (unsigned)(ga & 0xFFFFFFFFull);             // global_addr[31:0]
    g0.w = (unsigned)((ga >> 32) & 0x01FFFFFFull)      // global_addr[56:32]
         | (2u << 30);                                 // type=2 ("image")
    v8i g1;
    g1[0] = 0x00020000;                                // wg_mask=0, data_size=2 (4B)
    g1[1] = (int)((unsigned)(CHUNK_FLOATS & 0xFFFF) << 16); // tensor_dim0[15:0]
    g1[2] = 0x00010000;                                // tensor_dim0[31:16]=0, tensor_dim1=1
    g1[3] = (int)((unsigned)CHUNK_FLOATS << 16);       // dim1 hi=0, tile_dim0=8192
    g1[4] = 1;                                         // tile_dim1=1, tile_dim2=0
    g1[5] = CHUNK_FLOATS;                              // tensor_dim0_stride lo
    g1[6] = 0;                                         // stride hi, dim1_stride lo
    g1[7] = 0;
    v4i g2 = {0, 0, 0, 0};
    v4i g3 = {0, 0, 0, 0};
    v8i g4 = {0, 0, 0, 0, 0, 0, 0, 0};                 // 6-arg form (clang-23 lane)
    __builtin_amdgcn_tensor_load_to_lds(g0, g1, g2, g3, g4, 0);
  }
}
__device__ __forceinline__ void stage_wait(const float*, float*, int t) {
  if (t < 32) __builtin_amdgcn_s_wait_tensorcnt(0);
  __syncthreads();
}
#elif STAGE_MODE == 1
// ---- per-lane async copy path ----
typedef __attribute__((address_space(1))) void gas_void;
typedef __attribute__((address_space(3))) void las_void;
__device__ __forceinline__ void stage_issue(const float* gsrc, float* lds, int t) {
#pragma unroll
  for (int k = 0; k < 8; ++k) {
    const int idx = (t + k * 256) * 4;  // float4 granularity
    __builtin_amdgcn_global_load_async_to_lds_b128(
        (gas_void*)(unsigned long long)(gsrc + idx),
        (las_void*)(unsigned)(unsigned long long)(lds + idx), 0, 0);
  }
}
__device__ __forceinline__ void stage_wait(const float*, float*, int t) {
#if __has_builtin(__builtin_amdgcn_s_wait_asynccnt)
  __builtin_amdgcn_s_wait_asynccnt(0);
#else
  asm volatile("s_wait_asynccnt 0x0" ::: "memory");
#endif
  __syncthreads();
}
#else
// ---- fallback: synchronous staged copy ----
__device__ __forceinline__ void stage_issue(const float*, float*, int) {}
__device__ __forceinline__ void stage_wait(const float* gsrc, float* lds, int t) {
#pragma unroll
  for (int k = 0; k < 8; ++k) {
    const int idx = (t + k * 256) * 4;
    *(float4*)(lds + idx) = *(const float4*)(gsrc + idx);
  }
  __syncthreads();
}
#endif

// Reduce one 128-channel chunk (one head) held in LDS to mean/max per 4x4 patch.
__device__ __forceinline__ void reduce_chunk(const float* st, int t, int b, int head,
                                             float (*psum)[32], float (*pmax)[32],
                                             float* __restrict__ means,
                                             float* __restrict__ maxs) {
  const int pg   = t & 15;        // pixel-quad id: row h = pg>>1, cols (pg&1)*4..+3
  const int slot = t >> 4;        // 16 channel slots x 8 channels
  const int h    = pg >> 1;
  const int wq   = (pg & 1) << 2;
  const int ph   = h >> 1;
  const int pw2  = (pg & 1) << 1; // patch cols {pw2, pw2+1}
  float s0 = 0.f, s1 = 0.f;
  float m0 = -3.402823466e38f, m1 = -3.402823466e38f;
#pragma unroll
  for (int k = 0; k < 8; ++k) {
    const int c = slot * 8 + k;
    const float4 v = *(const float4*)(st + c * 64 + h * 8 + wq);
    s0 += v.x + v.y;
    s1 += v.z + v.w;
    m0 = fmaxf(m0, fmaxf(v.x, v.y));
    m1 = fmaxf(m1, fmaxf(v.z, v.w));
  }
  const int p0   = ph * 4 + pw2;
  const int pidx = slot * 2 + (h & 1);    // 32 partials per patch
  psum[p0][pidx] = s0; psum[p0 + 1][pidx] = s1;
  pmax[p0][pidx] = m0; pmax[p0 + 1][pidx] = m1;
  __syncthreads();
  if (t < 32) {
    const int patch = t & 15;
    if (t < 16) {
      float s = 0.f;
#pragma unroll
      for (int i = 0; i < 32; ++i) s += psum[patch][i];
      means[(b * 16 + head) * 16 + patch] = s * (1.f / 512.f);
    } else {
      float m = -3.402823466e38f;
#pragma unroll
      for (int i = 0; i < 32; ++i) m = fmaxf(m, pmax[patch][i]);
      maxs[(b * 16 + head) * 16 + patch] = m;
    }
  }
  __syncthreads();
}

// K1: 2048 WGs, each owns one (b, 256-channel slice) = two heads.
__global__ __launch_bounds__(256) void la_reduce(const float* __restrict__ x,
                                                 float* __restrict__ means,
                                                 float* __restrict__ maxs) {
  __shared__ float stage[2][CHUNK_FLOATS];  // 64 KiB double buffer
  __shared__ float psum[16][32];
  __shared__ float pmax[16][32];
  const int t = threadIdx.x;
  const int b = blockIdx.x >> 3;
  const int s = blockIdx.x & 7;
  const float* slab = x + ((size_t)b * 2048 + (size_t)s * 256) * 64;

  stage_issue(slab, stage[0], t);
  stage_wait(slab, stage[0], t);
  stage_issue(slab + CHUNK_FLOATS, stage[1], t);            // overlap with chunk-0 reduce
  reduce_chunk(stage[0], t, b, s * 2 + 0, psum, pmax, means, maxs);
  stage_wait(slab + CHUNK_FLOATS, stage[1], t);
  reduce_chunk(stage[1], t, b, s * 2 + 1, psum, pmax, means, maxs);
}

// K2: 4096 WGs, one per (b, head). Gate + elementwise apply.
__global__ __launch_bounds__(256) void la_apply(
    const float* __restrict__ x, const float* __restrict__ w1,
    const float* __restrict__ b1, const float* __restrict__ w2,
    const float* __restrict__ b2, const float* __restrict__ wv,
    const float* __restrict__ bv, const float* __restrict__ means,
    const float* __restrict__ maxs, float* __restrict__ out) {
  __shared__ float sm_means[256];
  __shared__ float sm_maxs[256];
  __shared__ float sm_w1[16];
  __shared__ float sm_g[16];
  __shared__ float sm_gate[64];
  __shared__ float sm_sc[6];
  const int t = threadIdx.x;
  const int b = blockIdx.x >> 4;
  const int head = blockIdx.x & 15;

  sm_means[t] = means[b * 256 + t];
  sm_maxs[t]  = maxs[b * 256 + t];
  if (t < 16) sm_w1[t] = w1[t];
  if (t == 0) {
    sm_sc[0] = b1[0]; sm_sc[1] = wv[0]; sm_sc[2] = wv[1];
    sm_sc[3] = bv[0]; sm_sc[4] = w2[head]; sm_sc[5] = b2[head];
  }
  __syncthreads();
  const float b1v = sm_sc[0], wv0 = sm_sc[1], wv1 = sm_sc[2];

  // f1 across 16 heads for all 16 patches; relu; wv-combine -> pre-affine gate g
#if __has_builtin(__builtin_amdgcn_wmma_f32_16x16x4_f32)
  if (t < 32) {  // wave 0, EXEC all ones (WMMA requirement)
    const int half = t >> 4;
    const int n = t & 15;
    v8f dm = {0.f, 0.f, 0.f, 0.f, 0.f, 0.f, 0.f, 0.f};
    v8f dx = {0.f, 0.f, 0.f, 0.f, 0.f, 0.f, 0.f, 0.f};
#pragma unroll
    for (int i = 0; i < 4; ++i) {
      // A 16x4 f32: lanes 0-15 hold K=0(v0),1(v1); lanes 16-31 hold K=2,3.
      // B 4x16 f32: v0 lanes0-15=K0/lanes16-31=K2; v1 = K1/K3 rows, N=lane%16.
      const int kA = 4 * i + (half ? 2 : 0);
      const int kB = 4 * i + (half ? 3 : 1);
      v2f a;  a.x  = sm_w1[kA];              a.y  = sm_w1[kB];
      v2f bm; bm.x = sm_means[kA * 16 + n];  bm.y = sm_means[kB * 16 + n];
      v2f bx; bx.x = sm_maxs[kA * 16 + n];   bx.y = sm_maxs[kB * 16 + n];
      dm = __builtin_amdgcn_wmma_f32_16x16x4_f32(false, a, false, bm, (short)0, dm,
                                                 false, false);
      dx = __builtin_amdgcn_wmma_f32_16x16x4_f32(false, a, false, bx, (short)0, dx,
                                                 false, false);
    }
    if (t < 16) {  // D row M=0 lives in element 0 of lanes 0-15
      const float rm = fmaxf(dm[0] + b1v, 0.f);
      const float rx = fmaxf(dx[0] + b1v, 0.f);
      sm_g[t] = wv0 * rm + wv1 * rx;
    }
  }
#else
  if (t < 16) {
    float dm = 0.f, dx = 0.f;
#pragma unroll
    for (int hh = 0; hh < 16; ++hh) {
      dm += sm_w1[hh] * sm_means[hh * 16 + t];
      dx += sm_w1[hh] * sm_maxs[hh * 16 + t];
    }
    sm_g[t] = wv0 * fmaxf(dm + b1v, 0.f) + wv1 * fmaxf(dx + b1v, 0.f);
  }
#endif
  __syncthreads();

  // Bilinear 4x4 -> 8x8 (half-pixel centers), affine by w2/b2/bv, sigmoid, +1.
  if (t < 64) {
    const int i = t >> 3, j = t & 7;
    const float ci = 0.5f * (float)i - 0.25f;
    const float cj = 0.5f * (float)j - 0.25f;
    const int fi0 = (int)floorf(ci);
    const int fj0 = (int)floorf(cj);
    const float fi = ci - (float)fi0;
    const float fj = cj - (float)fj0;
    const int il = fi0 < 0 ? 0 : fi0;
    const int ih = fi0 + 1 > 3 ? 3 : fi0 + 1;
    const int jl = fj0 < 0 ? 0 : fj0;
    const int jh = fj0 + 1 > 3 ? 3 : fj0 + 1;
    const float gu = sm_g[il * 4 + jl] * (1.f - fi) * (1.f - fj)
                   + sm_g[il * 4 + jh] * (1.f - fi) * fj
                   + sm_g[ih * 4 + jl] * fi * (1.f - fj)
                   + sm_g[ih * 4 + jh] * fi * fj;
    const float fv = gu * sm_sc[4] + (wv0 + wv1) * sm_sc[5] + sm_sc[3];
    sm_gate[t] = 1.f + 1.f / (1.f + expf(-fv));
  }
  __syncthreads();

  // out = x * (1 + sigmoid(gate)), float4 streaming (x re-read from L2).
  const float* xp = x + ((size_t)b * 2048 + (size_t)head * 128) * 64;
  float* op = out + ((size_t)b * 2048 + (size_t)head * 128) * 64;
  const int po = (t & 15) << 2;
  const float ga = sm_gate[po + 0], gb = sm_gate[po + 1];
  const float gc = sm_gate[po + 2], gd = sm_gate[po + 3];
#pragma unroll
  for (int k = 0; k < 8; ++k) {
    const int c = (t >> 4) + k * 16;
    float4 v = *(const float4*)(xp + c * 64 + po);
    v.x *= ga; v.y *= gb; v.z *= gc; v.w *= gd;
    *(float4*)(op + c * 64 + po) = v;
  }
}

extern "C" void kernel_launch(void* const* d_in, const int* in_sizes, int n_in,
                              void* d_out, int out_size, void* d_ws, size_t ws_size,
                              hipStream_t stream) {
  (void)in_sizes; (void)n_in; (void)out_size; (void)ws_size;
  const float* x  = (const float*)d_in[0];
  const float* w1 = (const float*)d_in[1];
  const float* b1 = (const float*)d_in[2];
  const float* w2 = (const float*)d_in[3];
  const float* b2 = (const float*)d_in[4];
  const float* wv = (const float*)d_in[5];
  const float* bv = (const float*)d_in[6];
  float* means = (float*)d_ws;                 // [256][16][16]
  float* maxs  = means + 256 * 16 * 16;        // [256][16][16] (512 KiB total)
  la_reduce<<<2048, 256, 0, stream>>>(x, means, maxs);
  la_apply<<<4096, 256, 0, stream>>>(x, w1, b1, w2, b2, wv, bv, means, maxs,
                                     (float*)d_out);
}